// Histogram_Binning_45286135169514
// MI455X (gfx1250) — compile-verified
//
#include <hip/hip_runtime.h>

#define NUM_BINS 15
#define NUM_CLASSES 19
#define BATCH 8
#define HDIM 512
#define WDIM 512

static constexpr int HW      = HDIM * WDIM;          // 262144 = 2^18
static constexpr int NPIX    = BATCH * HW;           // 2097152
static constexpr int PIX_PER_THREAD = 4;
static constexpr int BLOCK   = 256;
static constexpr int TABSZ   = NUM_CLASSES * NUM_BINS; // 285

typedef float v4f __attribute__((ext_vector_type(4)));

#if defined(__HIP_DEVICE_COMPILE__)
typedef __attribute__((address_space(1))) int as1_int;   // global
typedef __attribute__((address_space(3))) int as3_int;   // LDS
#endif

__global__ __launch_bounds__(BLOCK) void
Histogram_Binning_45286135169514_kernel(const float* __restrict__ logits,
                                        const float* __restrict__ val_freqs,
                                        float* __restrict__ out)
{
    __shared__ float tab[TABSZ];
    const int tid = threadIdx.x;

    // ---- Stage calibration table into LDS via CDNA5 async global->LDS path ----
#if defined(__HIP_DEVICE_COMPILE__) && \
    __has_builtin(__builtin_amdgcn_global_load_async_to_lds_b32) && \
    __has_builtin(__builtin_amdgcn_s_wait_asynccnt)
    for (int i = tid; i < TABSZ; i += BLOCK) {
        __builtin_amdgcn_global_load_async_to_lds_b32(
            (as1_int*)(val_freqs + i),
            (as3_int*)(&tab[i]),
            /*offset=*/0, /*cpol=*/0);
    }
    __builtin_amdgcn_s_wait_asynccnt(0);
#else
    for (int i = tid; i < TABSZ; i += BLOCK)
        tab[i] = val_freqs[i];
#endif
    __syncthreads();

    // ---- Each thread owns 4 consecutive pixels (same batch, consecutive w) ----
    const unsigned p0   = (unsigned)(blockIdx.x * BLOCK + tid) * PIX_PER_THREAD;
    const unsigned b    = p0 >> 18;            // / HW
    const unsigned hw   = p0 & (HW - 1);       // % HW
    const unsigned base = b * ((unsigned)NUM_CLASSES << 18) + hw;

    // Load all 19 class planes: B128 non-temporal, fully coalesced.
    v4f x[NUM_CLASSES];
#pragma unroll
    for (int c = 0; c < NUM_CLASSES; ++c)
        x[c] = __builtin_nontemporal_load(
                   (const v4f*)(logits + base + ((unsigned)c << 18)));

    // Row max (numerical stability, matches jax softmax)
    v4f m = x[0];
#pragma unroll
    for (int c = 1; c < NUM_CLASSES; ++c)
#pragma unroll
        for (int k = 0; k < 4; ++k)
            m[k] = fmaxf(m[k], x[c][k]);

    // exp + sum
    v4f s = 0.0f;
#pragma unroll
    for (int c = 0; c < NUM_CLASSES; ++c)
#pragma unroll
        for (int k = 0; k < 4; ++k) {
            x[c][k] = __expf(x[c][k] - m[k]);   // v_exp_f32 path
            s[k]   += x[c][k];
        }

    // t = NUM_BINS / sum  (one precise divide per pixel; folds the *15 mul)
    v4f t;
#pragma unroll
    for (int k = 0; k < 4; ++k)
        t[k] = (float)NUM_BINS * (1.0f / s[k]);

    // bin, LDS gather, class-sum
    v4f cs = 0.0f;
#pragma unroll
    for (int c = 0; c < NUM_CLASSES; ++c)
#pragma unroll
        for (int k = 0; k < 4; ++k) {
            int bi = (int)(x[c][k] * t[k]);     // floor(p*15) for p >= 0
            bi = bi > (NUM_BINS - 1) ? (NUM_BINS - 1) : (bi < 0 ? 0 : bi);
            float cal = tab[c * NUM_BINS + bi];
            x[c][k] = cal;
            cs[k]  += cal;
        }

    // zero-guarded normalization
#pragma unroll
    for (int k = 0; k < 4; ++k) {
        float d = (cs[k] == 0.0f) ? 1.0f : cs[k];
        cs[k] = 1.0f / d;
    }

    // Store 19 class planes: B128 non-temporal
#pragma unroll
    for (int c = 0; c < NUM_CLASSES; ++c) {
        v4f o;
#pragma unroll
        for (int k = 0; k < 4; ++k)
            o[k] = x[c][k] * cs[k];
        __builtin_nontemporal_store(o, (v4f*)(out + base + ((unsigned)c << 18)));
    }
}

extern "C" void kernel_launch(void* const* d_in, const int* in_sizes, int n_in,
                              void* d_out, int out_size, void* d_ws, size_t ws_size,
                              hipStream_t stream) {
    const float* logits    = (const float*)d_in[0];
    const float* val_freqs = (const float*)d_in[1];
    float*       out       = (float*)d_out;

    const int threads_total = NPIX / PIX_PER_THREAD;  // 524288
    const int grid = threads_total / BLOCK;           // 2048
    Histogram_Binning_45286135169514_kernel<<<grid, BLOCK, 0, stream>>>(
        logits, val_freqs, out);
}